// KnowledgeGraphGNN_21672404975688
// MI455X (gfx1250) — compile-verified
//
#include <hip/hip_runtime.h>

typedef __attribute__((ext_vector_type(2))) float v2f;
typedef __attribute__((ext_vector_type(8))) float v8f;

#define HDIM 128
#define GNUM 64
#define BN_EPS 1e-5f

// ---------------------------------------------------------------------------
// Degree / normalization kernels
// ---------------------------------------------------------------------------
__global__ void init_deg_kernel(float* deg, int n) {
    int i = blockIdx.x * blockDim.x + threadIdx.x;
    if (i < n) deg[i] = 1.0f;  // self-loop contributes 1
}

__global__ void count_deg_kernel(const int* __restrict__ dst, float* deg, int e) {
    int i = blockIdx.x * blockDim.x + threadIdx.x;
    if (i < e) atomicAdd(&deg[dst[i]], 1.0f);
}

__global__ void dinv_kernel(float* deg_dinv, int n) {
    int i = blockIdx.x * blockDim.x + threadIdx.x;
    if (i < n) {
        float d = deg_dinv[i];
        deg_dinv[i] = d > 0.0f ? rsqrtf(d) : 0.0f;
    }
}

__global__ void norm_kernel(const int* __restrict__ src, const int* __restrict__ dst,
                            const float* __restrict__ dinv, float* __restrict__ nrm, int e) {
    int i = blockIdx.x * blockDim.x + threadIdx.x;
    if (i < e) nrm[i] = dinv[src[i]] * dinv[dst[i]];
}

// ---------------------------------------------------------------------------
// fp32 WMMA GEMM:  C[M,Nc] = A[M,K] @ B[K,Nc]   (+ optional bias & relu)
// One wave (32 lanes) per 16-row stripe. A tile staged in LDS via
// GLOBAL_LOAD_ASYNC_TO_LDS_B128 (memory -> LDS, no VGPR round-trip, ASYNCcnt).
// LDS rows padded to K+4 floats: keeps every row 16B-aligned for B128 writes
// and gives conflict-free banks for the A-fragment reads.
// M, Nc multiples of 16; K multiple of 4.
// fuse: 0 = plain store, 1 = bias + relu
// ---------------------------------------------------------------------------
__global__ void wmma_gemm_f32(const float* __restrict__ A, const float* __restrict__ B,
                              const float* __restrict__ bias, float* __restrict__ C,
                              int M, int K, int Nc, int fuse) {
    extern __shared__ float ldsA[];            // 16 x (K+4) floats, padded stride
    const int lane = threadIdx.x;              // 0..31 (wave32)
    const int row0 = blockIdx.x * 16;
    const int ldk  = K + 4;                    // row stride: 16B-aligned, no bank conflicts

    // Async fill of the 16xK A tile: one float4 per lane per iteration,
    // consecutive lanes -> consecutive 16B chunks (coalesced), direct to LDS.
    const int kq = K >> 2;                     // float4s per row (trip count uniform: 16*kq % 32 == 0)
    for (int t = lane; t < 16 * kq; t += 32) {
        const int r  = t / kq;
        const int c4 = t % kq;
        unsigned long long gaddr =
            (unsigned long long)(uintptr_t)(A + (size_t)(row0 + r) * K + (c4 << 2));
        unsigned int laddr =
            (unsigned int)(uintptr_t)(&ldsA[r * ldk + (c4 << 2)]);
        asm volatile("global_load_async_to_lds_b128 %0, %1, off"
                     :: "v"(laddr), "v"(gaddr) : "memory");
    }
    asm volatile("s_wait_asynccnt 0" ::: "memory");
    __syncthreads();

    const int ln = lane & 15;                  // row (A) / col (B,D) within tile
    const int kk = (lane >> 4) << 1;           // K sub-offset: 0 or 2

    for (int nt = 0; nt < Nc; nt += 16) {
        v8f acc = {};
        for (int k = 0; k < K; k += 4) {
            v2f a, b;
            // A 16x4 fp32 fragment: lanes 0-15 hold K=k..k+1, lanes 16-31 K=k+2..k+3
            a.x = ldsA[ln * ldk + k + kk];
            a.y = ldsA[ln * ldk + k + kk + 1];
            // B 4x16 fp32 fragment (row striped across lanes); W is L2/WGP$-resident
            b.x = B[(size_t)(k + kk)     * Nc + nt + ln];
            b.y = B[(size_t)(k + kk + 1) * Nc + nt + ln];
            acc = __builtin_amdgcn_wmma_f32_16x16x4_f32(
                false, a, false, b, (short)0, acc, false, false);
        }
        // D 16x16 fp32: VGPR r -> M = (lane>=16 ? 8 : 0) + r, N = lane&15
        const int n  = nt + ln;
        const int mb = (lane >> 4) << 3;
        #pragma unroll
        for (int r = 0; r < 8; ++r) {
            float v = acc[r];
            if (fuse == 1) { v += bias[n]; v = v > 0.0f ? v : 0.0f; }
            C[(size_t)(row0 + mb + r) * Nc + n] = v;
        }
    }
}

// ---------------------------------------------------------------------------
// GCN aggregation pieces
// ---------------------------------------------------------------------------
// agg[i,c] = hw[i,c] * dinv[i]^2   (self-loop message; initializes accumulator)
__global__ void selfloop_init_kernel(const float* __restrict__ hw,
                                     const float* __restrict__ dinv,
                                     float* __restrict__ agg, int n) {
    long long idx = (long long)blockIdx.x * blockDim.x + threadIdx.x;
    if (idx >= (long long)n * HDIM) return;
    int i = (int)(idx >> 7);
    float d = dinv[i];
    agg[idx] = hw[idx] * d * d;
}

// agg[dst,c] += hw[src,c] * norm[e]   — 128 consecutive threads per edge
__global__ void edge_agg_kernel(const float* __restrict__ hw,
                                const int* __restrict__ src, const int* __restrict__ dst,
                                const float* __restrict__ nrm,
                                float* __restrict__ agg, int e) {
    long long idx = (long long)blockIdx.x * blockDim.x + threadIdx.x;
    if (idx >= (long long)e * HDIM) return;
    int ei = (int)(idx >> 7);
    int c  = (int)(idx & (HDIM - 1));
    float w = nrm[ei];
    float v = hw[(size_t)src[ei] * HDIM + c] * w;
    atomicAdd(&agg[(size_t)dst[ei] * HDIM + c], v);
}

// in place: h = relu( ((h + b) - rm) * rsqrt(rv+eps) * g + beta )
__global__ void bn_relu_kernel(float* __restrict__ h, const float* __restrict__ b,
                               const float* __restrict__ g, const float* __restrict__ beta,
                               const float* __restrict__ rm, const float* __restrict__ rv,
                               int n) {
    long long idx = (long long)blockIdx.x * blockDim.x + threadIdx.x;
    if (idx >= (long long)n * HDIM) return;
    int c = (int)(idx & (HDIM - 1));
    float v = h[idx] + b[c];
    v = (v - rm[c]) * rsqrtf(rv[c] + BN_EPS) * g[c] + beta[c];
    h[idx] = v > 0.0f ? v : 0.0f;
}

// ---------------------------------------------------------------------------
// Global mean pool
// ---------------------------------------------------------------------------
__global__ void pool_zero_kernel(float* pooled, float* cnt) {
    int i = blockIdx.x * blockDim.x + threadIdx.x;
    if (i < GNUM * HDIM) pooled[i] = 0.0f;
    if (i < GNUM) cnt[i] = 0.0f;
}

__global__ void pool_accum_kernel(const float* __restrict__ h, const int* __restrict__ batch,
                                  float* pooled, float* cnt, int n) {
    long long idx = (long long)blockIdx.x * blockDim.x + threadIdx.x;
    if (idx >= (long long)n * HDIM) return;
    int i = (int)(idx >> 7);
    int c = (int)(idx & (HDIM - 1));
    int gph = batch[i];
    atomicAdd(&pooled[gph * HDIM + c], h[idx]);
    if (c == 0) atomicAdd(&cnt[gph], 1.0f);
}

__global__ void pool_div_kernel(const float* __restrict__ pooled, const float* __restrict__ cnt,
                                float* __restrict__ xg) {
    int idx = blockIdx.x * blockDim.x + threadIdx.x;
    if (idx >= GNUM * HDIM) return;
    int gph = idx >> 7;
    xg[idx] = pooled[idx] / fmaxf(cnt[gph], 1.0f);
}

// out[g] = dot(hid[g,:64], Wc2) + bc2
__global__ void final_fc_kernel(const float* __restrict__ hid, const float* __restrict__ Wc2,
                                const float* __restrict__ bc2, float* __restrict__ out) {
    int gph = blockIdx.x * blockDim.x + threadIdx.x;
    if (gph >= GNUM) return;
    float s = bc2[0];
    #pragma unroll
    for (int k = 0; k < 64; ++k) s += hid[gph * 64 + k] * Wc2[k];
    out[gph] = s;
}

// ---------------------------------------------------------------------------
// Launch
// ---------------------------------------------------------------------------
extern "C" void kernel_launch(void* const* d_in, const int* in_sizes, int n_in,
                              void* d_out, int out_size, void* d_ws, size_t ws_size,
                              hipStream_t stream) {
    const float* x    = (const float*)d_in[0];
    const int*   ei   = (const int*)d_in[1];
    const int*   batch= (const int*)d_in[2];
    const float* W1 = (const float*)d_in[3];  const float* b1 = (const float*)d_in[4];
    const float* W2 = (const float*)d_in[5];  const float* b2 = (const float*)d_in[6];
    const float* W3 = (const float*)d_in[7];  const float* b3 = (const float*)d_in[8];
    const float* g1 = (const float*)d_in[9];  const float* be1= (const float*)d_in[10];
    const float* rm1= (const float*)d_in[11]; const float* rv1= (const float*)d_in[12];
    const float* g2 = (const float*)d_in[13]; const float* be2= (const float*)d_in[14];
    const float* rm2= (const float*)d_in[15]; const float* rv2= (const float*)d_in[16];
    const float* g3 = (const float*)d_in[17]; const float* be3= (const float*)d_in[18];
    const float* rm3= (const float*)d_in[19]; const float* rv3= (const float*)d_in[20];
    const float* We = (const float*)d_in[21]; const float* bev= (const float*)d_in[22];
    const float* Wc1= (const float*)d_in[23]; const float* bc1= (const float*)d_in[24];
    const float* Wc2= (const float*)d_in[25]; const float* bc2= (const float*)d_in[26];

    const int N = in_sizes[2];          // batch vector length == #nodes
    const int E = in_sizes[1] / 2;      // edge_index is [2,E]
    const int CIN = in_sizes[0] / N;    // 32
    const int* src = ei;
    const int* dst = ei + E;

    // workspace carve-out
    float* ws = (float*)d_ws;
    float* dinv   = ws;  ws += N;                  // deg -> dinv (in place)
    float* nrm    = ws;  ws += E;
    float* bufA   = ws;  ws += (size_t)N * HDIM;   // h / agg
    float* bufB   = ws;  ws += (size_t)N * HDIM;   // hw
    float* pooled = ws;  ws += GNUM * HDIM;
    float* cnt    = ws;  ws += GNUM;
    float* xg     = ws;  ws += GNUM * HDIM;
    float* emb    = ws;  ws += GNUM * HDIM;
    float* hid    = ws;  ws += GNUM * 64;

    const int TB = 256;
    const long long NC = (long long)N * HDIM;
    const long long EC = (long long)E * HDIM;
    const int gridNC = (int)((NC + TB - 1) / TB);
    const int gridEC = (int)((EC + TB - 1) / TB);

    // degrees & symmetric normalization (computed once, reused by all layers)
    init_deg_kernel<<<(N + TB - 1) / TB, TB, 0, stream>>>(dinv, N);
    count_deg_kernel<<<(E + TB - 1) / TB, TB, 0, stream>>>(dst, dinv, E);
    dinv_kernel<<<(N + TB - 1) / TB, TB, 0, stream>>>(dinv, N);
    norm_kernel<<<(E + TB - 1) / TB, TB, 0, stream>>>(src, dst, dinv, nrm, E);

    const int mTiles = N / 16;          // N = 100000 = 6250 * 16
    const size_t lds1 = (size_t)16 * (CIN  + 4) * sizeof(float);
    const size_t ldsH = (size_t)16 * (HDIM + 4) * sizeof(float);

    // ----- GCN layer 1 (K = CIN = 32) -----
    wmma_gemm_f32<<<mTiles, 32, lds1, stream>>>(x, W1, nullptr, bufB, N, CIN, HDIM, 0);
    selfloop_init_kernel<<<gridNC, TB, 0, stream>>>(bufB, dinv, bufA, N);
    edge_agg_kernel<<<gridEC, TB, 0, stream>>>(bufB, src, dst, nrm, bufA, E);
    bn_relu_kernel<<<gridNC, TB, 0, stream>>>(bufA, b1, g1, be1, rm1, rv1, N);

    // ----- GCN layer 2 (K = 128) -----
    wmma_gemm_f32<<<mTiles, 32, ldsH, stream>>>(bufA, W2, nullptr, bufB, N, HDIM, HDIM, 0);
    selfloop_init_kernel<<<gridNC, TB, 0, stream>>>(bufB, dinv, bufA, N);
    edge_agg_kernel<<<gridEC, TB, 0, stream>>>(bufB, src, dst, nrm, bufA, E);
    bn_relu_kernel<<<gridNC, TB, 0, stream>>>(bufA, b2, g2, be2, rm2, rv2, N);

    // ----- GCN layer 3 (K = 128) -----
    wmma_gemm_f32<<<mTiles, 32, ldsH, stream>>>(bufA, W3, nullptr, bufB, N, HDIM, HDIM, 0);
    selfloop_init_kernel<<<gridNC, TB, 0, stream>>>(bufB, dinv, bufA, N);
    edge_agg_kernel<<<gridEC, TB, 0, stream>>>(bufB, src, dst, nrm, bufA, E);
    bn_relu_kernel<<<gridNC, TB, 0, stream>>>(bufA, b3, g3, be3, rm3, rv3, N);

    // ----- global mean pool -----
    pool_zero_kernel<<<(GNUM * HDIM + TB - 1) / TB, TB, 0, stream>>>(pooled, cnt);
    pool_accum_kernel<<<gridNC, TB, 0, stream>>>(bufA, batch, pooled, cnt, N);
    pool_div_kernel<<<(GNUM * HDIM + TB - 1) / TB, TB, 0, stream>>>(pooled, cnt, xg);

    // ----- head: emb = relu(xg@We+be); hid = relu(emb@Wc1+bc1); out = hid@Wc2+bc2
    wmma_gemm_f32<<<GNUM / 16, 32, ldsH, stream>>>(xg, We, bev, emb, GNUM, HDIM, HDIM, 1);
    wmma_gemm_f32<<<GNUM / 16, 32, ldsH, stream>>>(emb, Wc1, bc1, hid, GNUM, HDIM, 64, 1);
    final_fc_kernel<<<1, GNUM, 0, stream>>>(hid, Wc2, bc2, (float*)d_out);
}